// BlockSparseLocallyConnected_31645319036953
// MI455X (gfx1250) — compile-verified
//
#include <hip/hip_runtime.h>

// Locally-connected layer as block-sparse GEMM on CDNA5 WMMA f32 16x16x4.
// out[b, n] = sum_k patch[b,n,k] * w[n,k] + bias[n]
// M = 16 batches, N = 16 windows (4x4 group), K = 25 tiles x 256 pixels.
// 4 waves/block; wave wv owns tile rows rr = 4*wv .. 4*wv+3 (16 K-chunks).

typedef __attribute__((ext_vector_type(2))) float v2f;
typedef __attribute__((ext_vector_type(4))) float v4f;
typedef __attribute__((ext_vector_type(8))) float v8f;

#define IMG_W   512
#define NKW_    32

__global__ __launch_bounds__(128) void lc_wmma_kernel(
    const float* __restrict__ x,      // (64, 1, 512, 512)
    const float* __restrict__ w,      // (1024, 1024)
    const float* __restrict__ bias,   // (1024)
    float* __restrict__ out)          // (64, 32, 32)
{
  __shared__ float xs[16 * 260];      // 16 batches x 256-pixel tile, padded stride
  __shared__ float red[4 * 256];      // per-wave partial accumulators

  const int tid  = threadIdx.x;
  const int wv   = __builtin_amdgcn_readfirstlane(tid >> 5);  // wave 0..3 (scalar)
  const int lane = tid & 31;

  const int wg  = blockIdx.x & 63;    // window group (8x8 groups of 4x4 windows)
  const int bg  = blockIdx.x >> 6;    // batch group (4 groups of 16)
  const int rw0 = (wg >> 3) * 4;
  const int cw0 = (wg & 7) * 4;
  const int b0  = bg * 16;

  // Per-lane WMMA fragment roles (ISA 16x16x4 f32 layouts):
  const int m    = lane & 15;         // A row  = batch within group
  const int hl   = lane >> 4;         // lane half -> K pair {0,1} or {2,3}
  const int koff = hl * 2;
  const int jw   = lane & 15;         // B col  = window within group
  const int jr   = jw >> 2, jc = jw & 3;

  // Wave-constant LDS base for A fragments (tile rows 4*wv..4*wv+3).
  const float* const abase = &xs[m * 260 + (wv * 4) * 16 + koff];

  v8f acc = {0.f, 0.f, 0.f, 0.f, 0.f, 0.f, 0.f, 0.f};

  for (int t = 0; t < 25; ++t) {
    const int ti = t / 5, tj = t % 5;
    const int tr = rw0 + ti;          // tile coords in padded image (0..32)
    const int tc = cw0 + tj;

    // ---- Stage tile: 16 batches x 16x16 f32 -> LDS, coalesced b128 loads.
    // pad=8 and W=512 are multiples of 4, so every float4 segment is fully
    // in-bounds or fully padding -> branchless clamp + select.
    #pragma unroll
    for (int it = 0; it < 8; ++it) {
      const int seg = tid + it * 128;     // 0..1023, uniform trip count
      const int sm = seg >> 6;            // batch 0..15
      const int rr = (seg >> 2) & 15;     // tile row
      const int cs = seg & 3;             // 4-float segment in row
      const int r  = tr * 16 + rr - 8;
      const int c0 = tc * 16 + cs * 4 - 8;
      const bool valid = ((unsigned)r < 512u) & ((unsigned)c0 <= 508u);
      const int rc = valid ? r : 0;
      const int cc = valid ? c0 : 0;
      v4f v = *(const v4f*)(x + (((size_t)(b0 + sm)) << 18)
                              + (size_t)rc * IMG_W + cc);
      const v4f z4 = {0.f, 0.f, 0.f, 0.f};
      v = valid ? v : z4;
      *(v4f*)(&xs[sm * 260 + rr * 16 + cs * 4]) = v;
    }
    __syncthreads();

    // ---- Per-lane weight pointer for this tile / window column.
    // Window (rw0+jr, cw0+jc) touches tile (ti,tj) iff a,b in {0,1}.
    const unsigned a  = (unsigned)(ti - jr);
    const unsigned bb = (unsigned)(tj - jc);
    const bool act = (a <= 1u) && (bb <= 1u);
    const int  n   = (rw0 + jr) * NKW_ + (cw0 + jc);
    const size_t woff = (size_t)n * 1024 + (a & 1u) * 512 + (bb & 1u) * 16
                      + koff + (size_t)(wv * 4) * 32;
    const float* wpt = act ? (w + woff) : w;   // safe dummy when inactive

    // ---- K chunks: 16 per wave, fully unrolled -> immediate offsets,
    // batched loads, staggered waits.
    #pragma unroll
    for (int i = 0; i < 16; ++i) {
      const int rr  = i >> 2;          // relative tile row 0..3
      const int cc0 = (i & 3) * 4;     // column segment
      v2f bwv = *(const v2f*)(wpt + rr * 32 + cc0);   // (K0,K1)/(K2,K3) pair
      const v2f z2 = {0.f, 0.f};
      bwv = act ? bwv : z2;                           // dead columns -> 0
      const v2f av = *(const v2f*)(abase + rr * 16 + cc0);
      acc = __builtin_amdgcn_wmma_f32_16x16x4_f32(
          false, av, false, bwv, (short)0, acc, false, false);
    }
    __syncthreads();
  }

  // ---- Deterministic fixed-order reduction of the 4 wave partials.
  #pragma unroll
  for (int v = 0; v < 8; ++v)
    red[wv * 256 + (v + 8 * hl) * 16 + (lane & 15)] = acc[v];
  __syncthreads();

  #pragma unroll
  for (int it = 0; it < 2; ++it) {
    const int idx = tid + it * 128;   // 0..255, uniform
    float s = red[idx] + red[256 + idx] + red[512 + idx] + red[768 + idx];
    const int M = idx >> 4, N = idx & 15;
    const int n = (rw0 + (N >> 2)) * NKW_ + (cw0 + (N & 3));
    out[((size_t)(b0 + M)) * 1024 + n] = s + bias[n];
  }
}

extern "C" void kernel_launch(void* const* d_in, const int* in_sizes, int n_in,
                              void* d_out, int out_size, void* d_ws, size_t ws_size,
                              hipStream_t stream) {
  const float* x    = (const float*)d_in[0];   // (64,1,512,512) f32
  const float* w    = (const float*)d_in[1];   // (1024,1024) f32
  const float* bias = (const float*)d_in[2];   // (1024,) f32
  float* out = (float*)d_out;                  // (64,32,32) f32
  // 64 window groups x 4 batch groups, 4 waves (128 threads) per block.
  lc_wmma_kernel<<<dim3(256), dim3(128), 0, stream>>>(x, w, bias, out);
}